// CircuitGNN_83683142795520
// MI455X (gfx1250) — compile-verified
//
#include <hip/hip_runtime.h>
#include <hip/hip_bf16.h>

typedef __attribute__((ext_vector_type(2))) float v2f;
typedef __attribute__((ext_vector_type(8))) float v8f;

#define NNODES 50000
#define NEDGES 600000
#define DH 128

// ---------------- degree / norm ----------------
__global__ void k_deg_init(float* deg, int n) {
    int i = blockIdx.x * blockDim.x + threadIdx.x;
    if (i < n) deg[i] = 1.0f;              // self-loop contributes 1
}

__global__ void k_deg_accum(float* deg, const int* __restrict__ dst, int e) {
    int i = blockIdx.x * blockDim.x + threadIdx.x;
    if (i < e) atomicAdd(&deg[dst[i]], 1.0f);
}

__global__ void k_dinv(float* deg, int n) {
    int i = blockIdx.x * blockDim.x + threadIdx.x;
    if (i < n) deg[i] = rsqrtf(deg[i]);    // deg >= 1 always (self-loops)
}

// ---------------- WMMA GEMM: out[N x FOUT] = (relu?)in[N x FIN] @ W (+bias) ----------------
// One wave computes TWO adjacent 16x16 tiles of out (shared A operand, 2 accumulators).
// blockDim.x = 32 * (FOUT/32). FIN/FOUT are compile-time so every global load uses a
// single base VGPR pair + immediate offset; K loop fully unrolled:
// 32 steps of v_wmma_f32_16x16x4_f32, 2 WMMAs per step.
template <bool RELU, bool BIAS, int FIN, int FOUT>
__global__ void k_gemm_wmma_f32(const float* __restrict__ A,
                                const float* __restrict__ W,
                                const float* __restrict__ bias,
                                float* __restrict__ out) {
    const int lane  = threadIdx.x & 31;
    const int wave  = threadIdx.x >> 5;
    const int m15   = lane & 15;
    const int hi    = lane >> 4;           // 0 for lanes 0-15, 1 for 16-31
    const int row   = blockIdx.x * 16 + m15;
    const int koff  = hi << 1;             // K offset within the 4-wide slab
    const int ncol0 = wave * 32 + m15;     // first N tile (second is +16 cols)

    const float* __restrict__ arow = A + (size_t)row * FIN + koff;
    const float* __restrict__ wp   = W + (size_t)koff * FOUT + ncol0;

    v8f acc0 = {};
    v8f acc1 = {};
    #pragma unroll
    for (int k0 = 0; k0 < FIN; k0 += 4) {
        // A operand: 16x4 f32, lanes 0-15 carry K=k0+{0,1}, lanes 16-31 K=k0+{2,3}
        float2 av = *(const float2*)(arow + k0);   // 8B aligned, immediate offset
        float a0 = av.x, a1 = av.y;
        if (RELU) { a0 = fmaxf(a0, 0.0f); a1 = fmaxf(a1, 0.0f); }
        v2f a; a[0] = a0; a[1] = a1;
        // B operands: 4x16 f32 tiles; all 4 fetches are base + compile-time immediate
        v2f b0, b1;
        b0[0] = wp[k0 * FOUT];
        b0[1] = wp[k0 * FOUT + FOUT];
        b1[0] = wp[k0 * FOUT + 16];
        b1[1] = wp[k0 * FOUT + FOUT + 16];
        acc0 = __builtin_amdgcn_wmma_f32_16x16x4_f32(
            false, a, false, b0, (short)0, acc0, false, false);
        acc1 = __builtin_amdgcn_wmma_f32_16x16x4_f32(
            false, a, false, b1, (short)0, acc1, false, false);
    }

    // C/D layout: VGPR r -> M = r (lanes 0-15) or r+8 (lanes 16-31), N = lane&15
    const int   wrow = blockIdx.x * 16 + (hi << 3);
    const float bv0  = BIAS ? bias[ncol0] : 0.0f;
    const float bv1  = BIAS ? bias[ncol0 + 16] : 0.0f;
    float* __restrict__ obase = out + (size_t)wrow * FOUT + ncol0;
    #pragma unroll
    for (int r = 0; r < 8; ++r) {
        obase[r * FOUT]      = acc0[r] + bv0;
        obase[r * FOUT + 16] = acc1[r] + bv1;
    }
}

// ---------------- agg init: bias + self-loop term (no atomics) ----------------
// agg[i][f] = b[f] + tmp[i][f] * dinv[i]^2 ; one lane per float4 group
__global__ void k_agg_init(const float* __restrict__ tmp,
                           const float* __restrict__ bias,
                           const float* __restrict__ dinv,
                           float* __restrict__ agg, int n) {
    int t = blockIdx.x * blockDim.x + threadIdx.x;
    int i = t >> 5;
    int g = t & 31;                         // which float4 of the 128 feats
    if (i >= n) return;
    float di = dinv[i];
    float w  = di * di;
    float4 v  = ((const float4*)(tmp + (size_t)i * DH))[g];
    float4 bb = ((const float4*)bias)[g];
    float4 r;
    r.x = bb.x + v.x * w;
    r.y = bb.y + v.y * w;
    r.z = bb.z + v.z * w;
    r.w = bb.w + v.w * w;
    ((float4*)(agg + (size_t)i * DH))[g] = r;
}

// ---------------- edge scatter: one wave32 per edge, float4 per lane ----------------
__global__ void k_edge_scatter(const float* __restrict__ tmp,
                               float* __restrict__ agg,
                               const int* __restrict__ src,
                               const int* __restrict__ dst,
                               const float* __restrict__ dinv, int e) {
    long long t = (long long)blockIdx.x * blockDim.x + threadIdx.x;
    int ei = (int)(t >> 5);
    int g  = (int)(t & 31);
    if (ei >= e) return;
    int s = src[ei], d = dst[ei];
    float w = dinv[s] * dinv[d];
    float4 v = ((const float4*)(tmp + (size_t)s * DH))[g];
    float* p = agg + (size_t)d * DH + (size_t)g * 4;
    atomicAdd(p + 0, v.x * w);
    atomicAdd(p + 1, v.y * w);
    atomicAdd(p + 2, v.z * w);
    atomicAdd(p + 3, v.w * w);
}

// ---------------- launch ----------------
extern "C" void kernel_launch(void* const* d_in, const int* in_sizes, int n_in,
                              void* d_out, int out_size, void* d_ws, size_t ws_size,
                              hipStream_t stream) {
    const float* x   = (const float*)d_in[0];
    const int*   ei  = (const int*)d_in[1];
    const float* W1  = (const float*)d_in[2];
    const float* b1  = (const float*)d_in[3];
    const float* W2  = (const float*)d_in[4];
    const float* b2  = (const float*)d_in[5];
    const float* W3  = (const float*)d_in[6];
    const float* b3  = (const float*)d_in[7];
    const float* Wl  = (const float*)d_in[8];
    const float* bl  = (const float*)d_in[9];

    const int* src = ei;            // edge_index[0]
    const int* dst = ei + NEDGES;   // edge_index[1]

    float* dinv = (float*)d_ws;                       // N floats
    float* tmp  = dinv + 50048;                       // N*128 floats (aligned pad)
    float* agg  = tmp + (size_t)NNODES * DH;          // N*128 floats

    // degrees -> dinv
    k_deg_init <<<(NNODES + 255) / 256, 256, 0, stream>>>(dinv, NNODES);
    k_deg_accum<<<(NEDGES + 255) / 256, 256, 0, stream>>>(dinv, dst, NEDGES);
    k_dinv     <<<(NNODES + 255) / 256, 256, 0, stream>>>(dinv, NNODES);

    const int mtiles    = NNODES / 16;                       // 3125, exact
    const int aggBlocks = (NNODES * 32 + 255) / 256;
    const int sctBlocks = (int)(((long long)NEDGES * 32 + 255) / 256);

    // layer 1: relu off (raw x input), bias folded into agg init
    k_gemm_wmma_f32<false, false, DH, DH><<<mtiles, 128, 0, stream>>>(x,   W1, nullptr, tmp);
    k_agg_init     <<<aggBlocks, 256, 0, stream>>>(tmp, b1, dinv, agg, NNODES);
    k_edge_scatter <<<sctBlocks, 256, 0, stream>>>(tmp, agg, src, dst, dinv, NEDGES);

    // layer 2: relu applied to agg inside GEMM A-load
    k_gemm_wmma_f32<true, false, DH, DH><<<mtiles, 128, 0, stream>>>(agg, W2, nullptr, tmp);
    k_agg_init     <<<aggBlocks, 256, 0, stream>>>(tmp, b2, dinv, agg, NNODES);
    k_edge_scatter <<<sctBlocks, 256, 0, stream>>>(tmp, agg, src, dst, dinv, NEDGES);

    // layer 3
    k_gemm_wmma_f32<true, false, DH, DH><<<mtiles, 128, 0, stream>>>(agg, W3, nullptr, tmp);
    k_agg_init     <<<aggBlocks, 256, 0, stream>>>(tmp, b3, dinv, agg, NNODES);
    k_edge_scatter <<<sctBlocks, 256, 0, stream>>>(tmp, agg, src, dst, dinv, NEDGES);

    // final linear: FOUT=32 -> 1 wave/block (2 tiles), bias fused, relu on input
    k_gemm_wmma_f32<true, true, DH, 32><<<mtiles, 32, 0, stream>>>(agg, Wl, bl, (float*)d_out);
}